// MLP_40243843564094
// MI455X (gfx1250) — compile-verified
//
#include <hip/hip_runtime.h>
#include <hip/hip_bf16.h>
#include <math.h>

// ---------------------------------------------------------------------------
// Problem constants (reference: B=128, IN=512, H=512, CLIP=1, DELAY_STEPS=2)
// ---------------------------------------------------------------------------
#define BATCH 128
#define HDIM  512
#define BH    (BATCH * HDIM)

typedef __attribute__((ext_vector_type(2))) float v2f;
typedef __attribute__((ext_vector_type(8))) float v8f;

__device__ __forceinline__ float fast_tanh(float x) {
#if __has_builtin(__builtin_amdgcn_tanhf)
    return __builtin_amdgcn_tanhf(x);
#else
    return tanhf(x);
#endif
}

__device__ __forceinline__ float clamp1(float x) {
    return fminf(fmaxf(x, -1.0f), 1.0f);
}

__device__ __forceinline__ float wave_sum(float v) {
#pragma unroll
    for (int off = 16; off > 0; off >>= 1) v += __shfl_xor(v, off, 32);
    return v;
}

// ---------------------------------------------------------------------------
// Dense GEMM  Y = act(X @ W + bias),  X:[M,K] W:[K,N] row-major, act: 0=none 1=tanh
// One wave per 16x16 C tile using V_WMMA_F32_16X16X4_F32 (full fp32 precision).
// Launch: grid = (M/16)*(N/16)/8 blocks of 256 threads (8 waves/block).
// ---------------------------------------------------------------------------
__global__ __launch_bounds__(256) void gemm_wmma_f32(
    const float* __restrict__ X, const float* __restrict__ W,
    const float* __restrict__ bias, float* __restrict__ Y,
    int M, int N, int K, int act)
{
    const int wid  = threadIdx.x >> 5;
    const int lane = threadIdx.x & 31;
    const int ntn  = N >> 4;
    const int tile = blockIdx.x * 8 + wid;
    if (tile >= (M >> 4) * ntn) return;           // wave-uniform; EXEC stays full
    const int tm = tile / ntn;
    const int tn = tile - tm * ntn;

    const int hlf = lane >> 4;                    // 0: lanes 0-15, 1: lanes 16-31
    const int lr  = lane & 15;
    const int col = tn * 16 + lr;

    const float bv = bias ? bias[col] : 0.0f;

#if __has_builtin(__builtin_amdgcn_wmma_f32_16x16x4_f32)
    // ISA layouts (CDNA5 ISA):
    //  A 16x4 f32:  lane<16 holds {K=0,K=1} of row M=lane; lane>=16 holds {K=2,K=3}
    //  B 4x16 f32:  v0: lanes0-15 row K=0, lanes16-31 row K=2; v1: K=1 / K=3
    //  C/D 16x16:   vgpr r: lanes0-15 -> M=r, lanes16-31 -> M=r+8; N=lane%16
    v8f c;
#pragma unroll
    for (int r = 0; r < 8; ++r) c[r] = bv;

    const float* Xrow = X + (size_t)(tm * 16 + lr) * K + hlf * 2;
    const float* Wp   = W + (size_t)(hlf * 2) * N + col;

    for (int k0 = 0; k0 < K; k0 += 4) {
        v2f a, b;
        a.x = Xrow[k0];
        a.y = Xrow[k0 + 1];
        b.x = Wp[(size_t)k0 * N];
        b.y = Wp[(size_t)(k0 + 1) * N];
        c = __builtin_amdgcn_wmma_f32_16x16x4_f32(
                false, a, false, b, (short)0, c, false, false);
    }

#pragma unroll
    for (int r = 0; r < 8; ++r) {
        const int row = tm * 16 + r + hlf * 8;
        float v = c[r];
        if (act == 1) v = fast_tanh(v);
        Y[(size_t)row * N + col] = v;
    }
#else
    // Fallback (same launch geometry / same tile mapping), plain FMA loop.
    for (int r = 0; r < 8; ++r) {
        const int row = tm * 16 + r + hlf * 8;
        float acc = bv;
        for (int k = 0; k < K; ++k)
            acc = fmaf(X[(size_t)row * K + k], W[(size_t)k * N + col], acc);
        if (act == 1) acc = fast_tanh(acc);
        Y[(size_t)row * N + col] = acc;
    }
#endif
}

// ---------------------------------------------------------------------------
// The big streaming pass over pw [B,H,H].  One wave per (b,h) row of 512.
// Optionally: update  pw' = clamp(pw + nm[b]*heb*tanh(preh[b,h]*prevpost[b,i]))
//             store   pw_out = pw'
//             matvec  plastic[b,h] = sum_i alpha[h,i]*pw'[b,h,i]*post[b,i]
// Launch: grid = B*H/8 = 8192 blocks of 256 threads.
// ---------------------------------------------------------------------------
__global__ __launch_bounds__(256) void pw_pass(
    const float* __restrict__ pw_in, float* __restrict__ pw_out,
    const float* __restrict__ alpha,
    const float* __restrict__ preh, const float* __restrict__ prevpost,
    const float* __restrict__ nmvec, const float* __restrict__ heb_mult,
    const float* __restrict__ post, float* __restrict__ plastic_out,
    int doUpdate, int doStore, int doMatvec)
{
    const int wid  = threadIdx.x >> 5;
    const int lane = threadIdx.x & 31;
    const int bh   = blockIdx.x * 8 + wid;        // 0 .. B*H-1
    const int b    = bh >> 9;
    const int h    = bh & 511;

    const size_t rowoff = (size_t)bh * HDIM;
    const float4* pwr = (const float4*)(pw_in + rowoff);
    float4*       pww = (float4*)(pw_out + rowoff);
    const float4* al4 = (const float4*)(alpha + (size_t)h * HDIM);
    const float4* pp4 = (const float4*)(prevpost + (size_t)b * HDIM);
    const float4* po4 = (const float4*)(post + (size_t)b * HDIM);

    float ph = 0.0f, s = 0.0f;
    if (doUpdate) {
        ph = preh[(size_t)b * HDIM + h];
        s  = nmvec[b] * heb_mult[0];
    }

    float acc = 0.0f;
#pragma unroll
    for (int c = 0; c < 4; ++c) {
        const int i4 = c * 32 + lane;             // coalesced float4 per lane
        float4 v = pwr[i4];
        if (doUpdate) {
            const float4 pp = pp4[i4];
            v.x = clamp1(v.x + s * fast_tanh(ph * pp.x));
            v.y = clamp1(v.y + s * fast_tanh(ph * pp.y));
            v.z = clamp1(v.z + s * fast_tanh(ph * pp.z));
            v.w = clamp1(v.w + s * fast_tanh(ph * pp.w));
        }
        if (doStore) pww[i4] = v;
        if (doMatvec) {
            const float4 a = al4[i4];
            const float4 p = po4[i4];
            acc = fmaf(a.x * v.x, p.x, acc);
            acc = fmaf(a.y * v.y, p.y, acc);
            acc = fmaf(a.z * v.z, p.z, acc);
            acc = fmaf(a.w * v.w, p.w, acc);
        }
    }
    if (doMatvec) {
        acc = wave_sum(acc);
        if (lane == 0) plastic_out[bh] = acc;
    }
}

// ---------------------------------------------------------------------------
// Step-0 head: pre = innate+plastic; hidden = tanh(pre);
// choice = sigmoid(hidden@choice_W + b); value = hidden@val_W + b;
// pre_hrc = hidden + reward*rew_W + rew_b.   One wave per batch row.
// ---------------------------------------------------------------------------
__global__ __launch_bounds__(32) void head_kernel(
    const float* __restrict__ innate, const float* __restrict__ plastic,
    const float* __restrict__ reward,
    const float* __restrict__ choice_W, const float* __restrict__ choice_b,
    const float* __restrict__ val_W, const float* __restrict__ val_b,
    const float* __restrict__ rew_W, const float* __restrict__ rew_b,
    float* __restrict__ pre, float* __restrict__ hidden_out,
    float* __restrict__ pre_hrc,
    float* __restrict__ out_choice, float* __restrict__ out_value)
{
    const int b = blockIdx.x;
    const int lane = threadIdx.x;
    const float rw = reward[b];
    float accC = 0.0f, accV = 0.0f;
    for (int i = lane; i < HDIM; i += 32) {
        const size_t idx = (size_t)b * HDIM + i;
        const float p = innate[idx] + plastic[idx];
        const float hd = fast_tanh(p);
        pre[idx] = p;
        hidden_out[idx] = hd;
        pre_hrc[idx] = hd + rw * rew_W[i] + rew_b[i];
        accC = fmaf(hd, choice_W[i], accC);
        accV = fmaf(hd, val_W[i], accV);
    }
    accC = wave_sum(accC);
    accV = wave_sum(accV);
    if (lane == 0) {
        out_choice[b] = 1.0f / (1.0f + expf(-(accC + choice_b[0])));
        out_value[b]  = accV + val_b[0];
    }
}

// ---------------------------------------------------------------------------
// Neuromod: nm = tanh(h@nm_W + nm_b); out = nm_mult*(nm0-nm1). One wave / batch.
// ---------------------------------------------------------------------------
__global__ __launch_bounds__(32) void nm_kernel(
    const float* __restrict__ hvec, const float* __restrict__ nm_W,
    const float* __restrict__ nm_b, const float* __restrict__ nm_mult,
    float* __restrict__ nm_out, float* __restrict__ out_nm)
{
    const int b = blockIdx.x;
    const int lane = threadIdx.x;
    float a0 = 0.0f, a1 = 0.0f;
    for (int i = lane; i < HDIM; i += 32) {
        const float hd = hvec[(size_t)b * HDIM + i];
        a0 = fmaf(hd, nm_W[2 * i + 0], a0);
        a1 = fmaf(hd, nm_W[2 * i + 1], a1);
    }
    a0 = wave_sum(a0);
    a1 = wave_sum(a1);
    if (lane == 0) {
        const float n0 = fast_tanh(a0 + nm_b[0]);
        const float n1 = fast_tanh(a1 + nm_b[1]);
        const float v = nm_mult[0] * (n0 - n1);
        nm_out[b] = v;
        if (out_nm) out_nm[b] = v;
    }
}

// ---------------------------------------------------------------------------
// pre_d = innate + plastic; hid_d = tanh(pre_d)  (elementwise over B*H)
// ---------------------------------------------------------------------------
__global__ __launch_bounds__(256) void addtanh_kernel(
    const float* __restrict__ innate, const float* __restrict__ plastic,
    float* __restrict__ pre_d, float* __restrict__ hid_d, int n)
{
    const int i = blockIdx.x * 256 + threadIdx.x;
    if (i < n) {
        const float p = innate[i] + plastic[i];
        pre_d[i] = p;
        hid_d[i] = fast_tanh(p);
    }
}

// ---------------------------------------------------------------------------
// Host-side orchestration
// ---------------------------------------------------------------------------
extern "C" void kernel_launch(void* const* d_in, const int* in_sizes, int n_in,
                              void* d_out, int out_size, void* d_ws, size_t ws_size,
                              hipStream_t stream)
{
    const float* items    = (const float*)d_in[0];
    const float* pw0      = (const float*)d_in[1];
    const float* reward   = (const float*)d_in[2];
    const float* emb_W    = (const float*)d_in[3];
    const float* emb_b    = (const float*)d_in[4];
    const float* ex_W     = (const float*)d_in[5];
    const float* ex_b     = (const float*)d_in[6];
    const float* fc2_W    = (const float*)d_in[7];
    const float* fc2_b    = (const float*)d_in[8];
    const float* hrc_W    = (const float*)d_in[9];
    const float* hrc_b    = (const float*)d_in[10];
    const float* choice_W = (const float*)d_in[11];
    const float* choice_b = (const float*)d_in[12];
    const float* rew_W    = (const float*)d_in[13];
    const float* rew_b    = (const float*)d_in[14];
    const float* nm_W     = (const float*)d_in[15];
    const float* nm_b     = (const float*)d_in[16];
    const float* val_W    = (const float*)d_in[17];
    const float* val_b    = (const float*)d_in[18];
    const float* alpha    = (const float*)d_in[19];
    const float* nm_mult  = (const float*)d_in[20];
    const float* heb_mult = (const float*)d_in[21];

    float* out        = (float*)d_out;
    float* out_choice = out;                                   // [B]
    float* out_nm     = out + BATCH;                           // [B]
    float* out_value  = out + 2 * BATCH;                       // [B]
    float* out_pw     = out + 3 * BATCH;                       // [B,H,H]
    float* out_hidden = out_pw + (size_t)BATCH * HDIM * HDIM;  // [B,H]

    float* ws = (float*)d_ws;
    float* t1      = ws + 0 * (size_t)BH;
    float* emb     = ws + 1 * (size_t)BH;
    float* innate  = ws + 2 * (size_t)BH;
    float* plastic = ws + 3 * (size_t)BH;
    float* pre     = ws + 4 * (size_t)BH;
    float* pre_hrc = ws + 5 * (size_t)BH;
    float* hrc     = ws + 6 * (size_t)BH;
    float* pre_d0  = ws + 7 * (size_t)BH;
    float* hid_d0  = ws + 8 * (size_t)BH;
    float* pre_d1  = ws + 9 * (size_t)BH;
    float* hid_d1  = ws + 10 * (size_t)BH;
    float* nm0     = ws + 11 * (size_t)BH;
    float* nmd0    = nm0 + BATCH;
    float* nmd1    = nm0 + 2 * BATCH;

    const dim3 gGemm(32), bGemm(256);        // 256 tiles / 8 waves per block
    const dim3 gPw(BH / 8), bPw(256);        // one wave per (b,h) row
    const dim3 gRow(BATCH), bRow(32);        // one wave per batch row
    const dim3 gEw(BH / 256), bEw(256);

    // ---- step 0 forward ----
    gemm_wmma_f32<<<gGemm, bGemm, 0, stream>>>(items, emb_W, emb_b, t1, BATCH, HDIM, HDIM, 1);
    gemm_wmma_f32<<<gGemm, bGemm, 0, stream>>>(t1, ex_W, ex_b, emb, BATCH, HDIM, HDIM, 1);
    gemm_wmma_f32<<<gGemm, bGemm, 0, stream>>>(emb, fc2_W, fc2_b, innate, BATCH, HDIM, HDIM, 0);
    // A0: plastic = einsum(alpha*pw0, emb)
    pw_pass<<<gPw, bPw, 0, stream>>>(pw0, out_pw, alpha, pre, emb, nm0, heb_mult,
                                     emb, plastic, 0, 0, 1);
    head_kernel<<<gRow, bRow, 0, stream>>>(innate, plastic, reward,
                                           choice_W, choice_b, val_W, val_b,
                                           rew_W, rew_b, pre, out_hidden, pre_hrc,
                                           out_choice, out_value);
    gemm_wmma_f32<<<gGemm, bGemm, 0, stream>>>(pre_hrc, hrc_W, hrc_b, hrc, BATCH, HDIM, HDIM, 1);
    nm_kernel<<<gRow, bRow, 0, stream>>>(hrc, nm_W, nm_b, nm_mult, nm0, out_nm);

    // ---- fused B0 (pw0->pw1, trace=tanh(pre (x) emb)) + A1 (matvec with hrc) ----
    pw_pass<<<gPw, bPw, 0, stream>>>(pw0, out_pw, alpha, pre, emb, nm0, heb_mult,
                                     hrc, plastic, 1, 1, 1);

    // ---- delay step 0 ----
    gemm_wmma_f32<<<gGemm, bGemm, 0, stream>>>(hrc, fc2_W, fc2_b, innate, BATCH, HDIM, HDIM, 0);
    addtanh_kernel<<<gEw, bEw, 0, stream>>>(innate, plastic, pre_d0, hid_d0, BH);
    nm_kernel<<<gRow, bRow, 0, stream>>>(hid_d0, nm_W, nm_b, nm_mult, nmd0, nullptr);

    // ---- fused B1 (pw1->pw2, trace=tanh(pre_hrc (x) hidden)) + A2 (matvec with hid_d0) ----
    pw_pass<<<gPw, bPw, 0, stream>>>(out_pw, out_pw, alpha, pre_hrc, out_hidden, nmd0,
                                     heb_mult, hid_d0, plastic, 1, 1, 1);

    // ---- delay step 1 ----
    gemm_wmma_f32<<<gGemm, bGemm, 0, stream>>>(hid_d0, fc2_W, fc2_b, innate, BATCH, HDIM, HDIM, 0);
    addtanh_kernel<<<gEw, bEw, 0, stream>>>(innate, plastic, pre_d1, hid_d1, BH);
    nm_kernel<<<gRow, bRow, 0, stream>>>(hid_d1, nm_W, nm_b, nm_mult, nmd1, nullptr);

    // ---- final update B2 (pw2->pw3, trace=tanh(pre_d0 (x) hrc)), no matvec ----
    pw_pass<<<gPw, bPw, 0, stream>>>(out_pw, out_pw, alpha, pre_d0, hrc, nmd1,
                                     heb_mult, hrc, plastic, 1, 1, 0);
}